// TopLoss_4372276707641
// MI455X (gfx1250) — compile-verified
//
#include <hip/hip_runtime.h>
#include <math.h>

// ---------------------------------------------------------------------------
// TopLoss greedy-matching for MI455X (gfx1250, wave32).
//
//  Kernel 1 (wmma_d2): D2[i][j] = |p_i - g_j|^2 via V_WMMA_F32_16X16X4_F32.
//     A row i (K=4)  = [ px, py, |p|^2, 1 ]
//     B col j (K=4)  = [ -2gx, -2gy, 1, |g|^2 ]
//     A.B = |p|^2 + |g|^2 - 2 p.g = |p-g|^2.   One WMMA per 16x16 tile.
//  Kernel 2 (greedy_scan): persistent single workgroup (32 waves) running the
//     6000 sequential argmin rounds with u64-packed (dist,col) keys,
//     wave32 shfl reductions, owner-marking of matched columns in registers,
//     TDM (tensor_load_to_lds) staging of pred into LDS, and
//     global_prefetch of the next D2 row.
// ---------------------------------------------------------------------------

#define NPTS 6000          // pred points
#define MPTS 6000          // gt points
#define SCAN_THREADS 1024  // 32 wave32 waves, one WGP
#define CPT 8              // gt columns owned per thread (750*8 == 6000)

typedef __attribute__((ext_vector_type(2))) float v2f;
typedef __attribute__((ext_vector_type(8))) float v8f;
typedef __attribute__((ext_vector_type(4))) unsigned int u32x4;
typedef __attribute__((ext_vector_type(8))) unsigned int u32x8;

// ---------------------------------------------------------------------------
// Kernel 1: squared-distance matrix via f32 WMMA (16x16x4).
// One wave per 16x16 tile; 8 waves (256 threads) per block.
// ---------------------------------------------------------------------------
__global__ __launch_bounds__(256) void wmma_d2_kernel(
    const float* __restrict__ pred, const float* __restrict__ gt,
    float* __restrict__ D2) {
  const int TILES_N = MPTS / 16;  // 375
  const int TILES_M = NPTS / 16;  // 375
  int gwave = (int)(blockIdx.x * (blockDim.x / 32) + (threadIdx.x >> 5));
  int tile_m = gwave / TILES_N;
  int tile_n = gwave % TILES_N;
  if (tile_m >= TILES_M) return;  // whole-wave exit; live waves keep EXEC=all-1

  int lane = threadIdx.x & 31;
  int l16 = lane & 15;
  bool hi = lane >= 16;

  // A matrix (16x4, f32): lanes 0-15 hold K=0,1 in VGPR0,1; lanes 16-31 K=2,3.
  int m = tile_m * 16 + l16;
  float px = pred[2 * m + 0];
  float py = pred[2 * m + 1];
  v2f a;
  if (!hi) { a.x = px;               a.y = py;   }  // K=0 / K=1
  else     { a.x = px * px + py * py; a.y = 1.0f; } // K=2 / K=3

  // B matrix (4x16, f32): mirrored layout (lane = N, halves split K).
  int n = tile_n * 16 + l16;
  float gx = gt[2 * n + 0];
  float gy = gt[2 * n + 1];
  v2f b;
  if (!hi) { b.x = -2.0f * gx; b.y = -2.0f * gy;        }  // K=0 / K=1
  else     { b.x = 1.0f;       b.y = gx * gx + gy * gy; }  // K=2 / K=3

  v8f c = {};
  v8f d = __builtin_amdgcn_wmma_f32_16x16x4_f32(
      /*neg_a=*/false, a, /*neg_b=*/false, b,
      /*c_mod=*/(short)0, c, /*reuse_a=*/false, /*reuse_b=*/false);

  // D layout: VGPR r -> row (hi?8:0)+r, col = l16.
  int row0 = tile_m * 16 + (hi ? 8 : 0);
  int col = tile_n * 16 + l16;
#pragma unroll
  for (int r = 0; r < 8; ++r)
    D2[(size_t)(row0 + r) * MPTS + col] = d[r];
}

// ---------------------------------------------------------------------------
// Kernel 2: sequential greedy matching. Single workgroup, 32 waves.
// ---------------------------------------------------------------------------
__device__ __forceinline__ unsigned long long u64min(unsigned long long x,
                                                     unsigned long long y) {
  return x < y ? x : y;
}

template <bool USE_D2>
__global__ __launch_bounds__(SCAN_THREADS, 1) void greedy_scan_kernel(
    const float* __restrict__ pred, const float* __restrict__ gt,
    const float* __restrict__ D2, float* __restrict__ out) {
  const int tid = (int)threadIdx.x;
  const int lane = tid & 31;
  const int wid = tid >> 5;

  __shared__ float s_pred[2 * NPTS];                 // 48 KB (of 320 KB/WGP)
  __shared__ unsigned long long s_wkey[32];
  __shared__ unsigned long long s_win;

  // ---- Stage pred points into LDS with the Tensor Data Mover (wave 0). ----
  if (wid == 0) {
    unsigned long long ga = (unsigned long long)pred;
    // Low 32 bits of the flat LDS pointer == LDS byte offset (aperture rule).
    unsigned lds_off = (unsigned)(unsigned long long)(void*)s_pred;
    const unsigned NE = 2u * NPTS;  // 12000 f32 elements
    // D# group 0: count=1 | lds_addr | global_addr(57b) | type=2.
    u32x4 g0 = {1u, lds_off, (unsigned)ga,
                (unsigned)((ga >> 32) & 0x1FFFFFFull) | (2u << 30)};
    // D# group 1: wg_mask=0, data_size=2 (4B); tensor_dim0=tile_dim0=NE,
    // tensor_dim1=tile_dim1=1, stride0=NE. (1 x 12000 tile, no pad/iterate.)
    u32x8 g1 = {0x00020000u,
                (NE & 0xFFFFu) << 16,          // tensor_dim0[15:0]
                (1u << 16),                    // tensor_dim0 hi=0 | tensor_dim1 lo=1
                (NE & 0xFFFFu) << 16,          // tensor_dim1 hi=0 | tile_dim0=NE
                1u,                            // tile_dim1=1 | tile_dim2=0
                NE,                            // tensor_dim0_stride lo
                0u, 0u};
    asm volatile("tensor_load_to_lds %0, %1" ::"s"(g0), "s"(g1) : "memory");
    __builtin_amdgcn_s_wait_tensorcnt(0);
  }
  __syncthreads();

  // ---- Per-thread candidate state (owner-marking, no global used[] mask). --
  float gxr[CPT], gyr[CPT];
  unsigned alive = 0;
  if (!USE_D2) {
#pragma unroll
    for (int k = 0; k < CPT; ++k) {
      int col = tid * CPT + k;
      if (col < MPTS) {
        float2 g = ((const float2*)gt)[col];
        gxr[k] = g.x;
        gyr[k] = g.y;
      } else {
        gxr[k] = 3.0e18f;  // finite sentinel: d2 ~ 1.8e37 < FLT_MAX
        gyr[k] = 3.0e18f;
      }
    }
  } else {
    alive = (tid * CPT + CPT <= MPTS) ? 0xFFu : 0u;  // threads 0..749 active
  }

  float acc = 0.0f;  // running sum of matched squared distances (thread 0)

  for (int i = 0; i < NPTS; ++i) {
    unsigned long long key = ~0ull;

    if (USE_D2) {
      if (alive) {
        const float4* rp = (const float4*)(D2 + (size_t)i * MPTS + tid * CPT);
        float4 lo = rp[0], hi4 = rp[1];
        float v[CPT] = {lo.x, lo.y, lo.z, lo.w, hi4.x, hi4.y, hi4.z, hi4.w};
#pragma unroll
        for (int k = 0; k < CPT; ++k) {
          unsigned db =
              ((alive >> k) & 1u) ? __float_as_uint(v[k]) : 0x7F800000u;
          unsigned long long cand = ((unsigned long long)db << 32) |
                                    (unsigned)(tid * CPT + k);
          key = u64min(key, cand);
        }
        if (i + 1 < NPTS)  // hide next row's L2/HBM latency behind barriers
          __builtin_prefetch(D2 + (size_t)(i + 1) * MPTS + tid * CPT, 0, 0);
      }
    } else {
      float px = s_pred[2 * i + 0];
      float py = s_pred[2 * i + 1];
#pragma unroll
      for (int k = 0; k < CPT; ++k) {
        float dx = px - gxr[k];
        float dy = py - gyr[k];
        float d2 = fmaf(dx, dx, dy * dy);
        unsigned long long cand =
            ((unsigned long long)__float_as_uint(d2) << 32) |
            (unsigned)(tid * CPT + k);
        key = u64min(key, cand);
      }
    }

    // Wave32 argmin reduction (u64 keys: dist-major, smallest col on ties).
#pragma unroll
    for (int off = 16; off > 0; off >>= 1)
      key = u64min(key, (unsigned long long)__shfl_xor(key, off, 32));
    if (lane == 0) s_wkey[wid] = key;
    __syncthreads();

    if (wid == 0) {
      unsigned long long k2 = s_wkey[lane];  // exactly 32 wave results
#pragma unroll
      for (int off = 16; off > 0; off >>= 1)
        k2 = u64min(k2, (unsigned long long)__shfl_xor(k2, off, 32));
      if (lane == 0) {
        s_win = k2;
        acc += __uint_as_float((unsigned)(k2 >> 32));  // winning |p-g|^2
      }
    }
    __syncthreads();

    // Owner retires the matched gt column.
    unsigned wcol = (unsigned)s_win;
    if ((int)(wcol / CPT) == tid) {
      int k = (int)(wcol & (CPT - 1));
      if (USE_D2) {
        alive &= ~(1u << k);
      } else {
        gxr[k] = 3.0e18f;
        gyr[k] = 3.0e18f;
      }
    }
  }

  if (tid == 0) out[0] = sqrtf(acc);  // ||pred - matched||_F
}

// ---------------------------------------------------------------------------
extern "C" void kernel_launch(void* const* d_in, const int* in_sizes, int n_in,
                              void* d_out, int out_size, void* d_ws,
                              size_t ws_size, hipStream_t stream) {
  (void)in_sizes; (void)n_in; (void)out_size;
  const float* pred = (const float*)d_in[0];
  const float* gt = (const float*)d_in[1];
  float* out = (float*)d_out;

  const size_t need = (size_t)NPTS * (size_t)MPTS * sizeof(float);  // 144 MB
  if (ws_size >= need) {
    float* D2 = (float*)d_ws;
    const int total_waves = (NPTS / 16) * (MPTS / 16);  // 140625 tiles
    const int waves_per_block = 8;
    const int blocks = (total_waves + waves_per_block - 1) / waves_per_block;
    wmma_d2_kernel<<<blocks, waves_per_block * 32, 0, stream>>>(pred, gt, D2);
    greedy_scan_kernel<true><<<1, SCAN_THREADS, 0, stream>>>(pred, gt, D2, out);
  } else {
    greedy_scan_kernel<false><<<1, SCAN_THREADS, 0, stream>>>(pred, gt,
                                                              nullptr, out);
  }
}